// MetaCNN_46127948759100
// MI455X (gfx1250) — compile-verified
//
#include <hip/hip_runtime.h>
#include <hip/hip_bf16.h>

typedef float v2f __attribute__((ext_vector_type(2)));
typedef float v8f __attribute__((ext_vector_type(8)));

// ---------------------------------------------------------------------------
// Kernel 1: 5x5 SAME conv (3->64) + bias + ReLU.  feat[n][g][y][x]
// 4*64*128*128 = 4,194,304 outputs, 75 MACs each.
// ---------------------------------------------------------------------------
__global__ __launch_bounds__(256) void conv5_relu_kernel(
    const float* __restrict__ x, const float* __restrict__ w,
    const float* __restrict__ b, float* __restrict__ feat) {
  int idx = blockIdx.x * 256 + threadIdx.x;
  int xx = idx & 127;
  int y  = (idx >> 7) & 127;
  int g  = (idx >> 14) & 63;
  int n  = idx >> 20;
  float s = b[g];
  const float* wg = w + g * 75;         // [3][5][5]
  const float* xn = x + n * 3 * 16384;  // [3][128][128]
#pragma unroll
  for (int c = 0; c < 3; ++c) {
#pragma unroll
    for (int ki = 0; ki < 5; ++ki) {
      int yy = y + ki - 2;
      if (yy < 0 || yy > 127) continue;
#pragma unroll
      for (int kj = 0; kj < 5; ++kj) {
        int xj = xx + kj - 2;
        if (xj < 0 || xj > 127) continue;
        s = fmaf(xn[c * 16384 + yy * 128 + xj], wg[c * 25 + ki * 5 + kj], s);
      }
    }
  }
  feat[idx] = fmaxf(s, 0.0f);
}

// ---------------------------------------------------------------------------
// Fused main kernel: per workgroup = 16 consecutive upscaled pixels
// (fixed oy, ox = oxb..oxb+15).
//  phase 1: stage 3x3-unfolded feat patches (cols) + hidden MLP tile in LDS
//  phase 2: hidden[16,256] @ w2[256,1728] via V_WMMA_F32_16X16X4_F32,
//           consuming each 16x16 lw tile immediately against cols.
// ---------------------------------------------------------------------------

// Consumption with compile-time chunk phase M = (16*j) % 3 so that
// c = kc%3 / dk offsets fold to constants (6 register accumulators).
template <int M>
__device__ __forceinline__ void consume16(const float lw[16],
                                          const float* __restrict__ cA,
                                          const float* __restrict__ cB,
                                          int kstart,
                                          float (&a0)[3], float (&a1)[3]) {
#pragma unroll
  for (int q = 0; q < 16; ++q) {
    const int c  = (M + q) % 3;
    const int dk = (M + q) / 3;
    float l = lw[q];
    a0[c] = fmaf(l, cA[kstart + dk], a0[c]);
    a1[c] = fmaf(l, cB[kstart + dk], a1[c]);
  }
}

__global__ __launch_bounds__(128) void meta_upscale_kernel(
    const float* __restrict__ feat, const float* __restrict__ pos,
    const float* __restrict__ w1, const float* __restrict__ b1,
    const float* __restrict__ w2, const float* __restrict__ b2,
    float* __restrict__ out) {
  // LDS (96,000 B total; padded strides avoid 64-bank aliasing)
  __shared__ float lds_cols[4 * 8 * 580];   // [n][li][k<576], stride 580
  __shared__ float lds_hidden[16 * 260];    // [row][h<256], stride 260
  __shared__ float lds_lw[4 * 16 * 17];     // per-wave [row][kcL], stride 17
  __shared__ float lds_out[192];            // [row][n][c]

  const int tid = threadIdx.x;
  const int wg  = blockIdx.x;          // 4096 workgroups
  const int oy  = wg >> 4;             // 0..255
  const int oxb = (wg & 15) << 4;      // 0,16,...,240
  const int h0  = oy >> 1;             // source row
  const int wb  = oxb >> 1;            // source col base (8 distinct w)

  // --- phase 1a: zero out accumulator tile -------------------------------
  for (int e = tid; e < 192; e += 128) lds_out[e] = 0.0f;

  // --- phase 1b: stage unfold-3x3 cols: [4 n][8 li][576 k] ---------------
  for (int e = tid; e < 4 * 8 * 576; e += 128) {
    int n   = e / 4608;
    int rem = e - n * 4608;
    int li  = rem / 576;
    int k   = rem - li * 576;
    int c0  = k / 9;
    int kk  = k - c0 * 9;
    int ki  = kk / 3;
    int kj  = kk - ki * 3;
    int y   = h0 + ki - 1;
    int xc  = wb + li + kj - 1;
    float v = 0.0f;
    if (y >= 0 && y < 128 && xc >= 0 && xc < 128)
      v = feat[((n * 64 + c0) * 128 + y) * 128 + xc];
    lds_cols[(n * 8 + li) * 580 + k] = v;
  }

  // --- phase 1c: hidden tile: relu(pos @ w1 + b1), [16 row][256 h] -------
  for (int e = tid; e < 16 * 256; e += 128) {
    int row = e >> 8;
    int hh  = e & 255;
    int r   = oy * 256 + oxb + row;   // output-pixel index into pos_mat
    float p0 = pos[r * 3 + 0];
    float p1 = pos[r * 3 + 1];
    float p2 = pos[r * 3 + 2];
    float hv = fmaf(p0, w1[hh],
               fmaf(p1, w1[256 + hh],
               fmaf(p2, w1[512 + hh], b1[hh])));
    lds_hidden[row * 260 + hh] = fmaxf(hv, 0.0f);
  }

  __syncthreads();

  // --- phase 2: WMMA GEMM over 108 chunks of 16 columns (1728 = 108*16) --
  const int wv   = tid >> 5;
  const int lane = tid & 31;
  const int mrow = lane & 15;       // M row / N col position within tile
  const int half = lane >> 4;       // hi/lo half of wave
  const int li   = mrow >> 1;

  float acc0[3] = {0.f, 0.f, 0.f};  // n = half
  float acc1[3] = {0.f, 0.f, 0.f};  // n = half + 2
  const float* colsA = &lds_cols[((half)     * 8 + li) * 580];
  const float* colsB = &lds_cols[((half + 2) * 8 + li) * 580];
  float*       lwme  = &lds_lw[wv * 272];                 // this wave's tile
  const float* lwrow = &lds_lw[wv * 272 + mrow * 17];
  const float* hidp  = &lds_hidden[mrow * 260 + 2 * half];

  // Incrementally-maintained chunk state (chunks j = wv, wv+4, ..., <108):
  //   nb     = 16*j        (column base into w2; +64 per step)
  //   m3     = nb % 3      (+1 mod 3 per step, since 64 % 3 == 1)
  //   kstart = nb / 3      (+21 per step, +1 extra on m3 wrap)
  // Avoids per-chunk integer divisions / divergent-looking magic-mul code.
  int nb     = wv << 4;
  int m3     = (wv << 4) % 3;   // one-time division, outside the hot loop
  int kstart = (wv << 4) / 3;

  for (int t = 0; t < 27; ++t) {
    v8f acc = {0.f, 0.f, 0.f, 0.f, 0.f, 0.f, 0.f, 0.f};
    const float* w2p = w2 + nb + mrow;         // lane's output column

#pragma unroll 8
    for (int kt = 0; kt < 64; ++kt) {
      const int kb = kt << 2;
      // A frag (16x4 f32): lane holds M=mrow, K = kb + 2*half + {0,1}
      v2f a;
      a.x = hidp[kb];
      a.y = hidp[kb + 1];
      // B frag (4x16 f32): lane holds N=mrow, K = kb + 2*half + {0,1}
      v2f bvec;
      bvec.x = w2p[(kb + 2 * half)     * 1728];
      bvec.y = w2p[(kb + 2 * half + 1) * 1728];
      acc = __builtin_amdgcn_wmma_f32_16x16x4_f32(
          false, a, false, bvec, (short)0, acc, false, false);
    }

    // spill lw tile (+ b2 bias) to this wave's LDS region
    const float bias = b2[nb + mrow];
#pragma unroll
    for (int v = 0; v < 8; ++v) {
      int row = v + 8 * half;                  // C layout: M = v + 8*(t/16)
      lwme[row * 17 + mrow] = acc[v] + bias;
    }
    // same wave produced all 16x16 entries; DS ops are in-order per wave.

    float lw16[16];
#pragma unroll
    for (int q = 0; q < 16; ++q) lw16[q] = lwrow[q];

    if (m3 == 0)      consume16<0>(lw16, colsA, colsB, kstart, acc0, acc1);
    else if (m3 == 1) consume16<1>(lw16, colsA, colsB, kstart, acc0, acc1);
    else              consume16<2>(lw16, colsA, colsB, kstart, acc0, acc1);

    // advance chunk state: nb += 64
    nb     += 64;
    kstart += 21;
    m3     += 1;
    if (m3 == 3) { m3 = 0; kstart += 1; }
  }

  // --- merge the 4 waves' partial sums ----------------------------------
#pragma unroll
  for (int c = 0; c < 3; ++c) {
    atomicAdd(&lds_out[(mrow * 4 + half)     * 3 + c], acc0[c]);
    atomicAdd(&lds_out[(mrow * 4 + half + 2) * 3 + c], acc1[c]);
  }
  __syncthreads();

  // --- write out with MeanShift add (mean * 255) ------------------------
  for (int e = tid; e < 192; e += 128) {
    int row = e / 12;
    int rem = e - row * 12;
    int n = rem / 3;
    int c = rem - n * 3;
    float mv = (c == 0) ? 0.4488f * 255.0f
             : (c == 1) ? 0.4371f * 255.0f
                        : 0.4040f * 255.0f;
    int ox = oxb + row;
    out[((n * 3 + c) * 256 + oy) * 256 + ox] = lds_out[e] + mv;
  }
}

// ---------------------------------------------------------------------------
extern "C" void kernel_launch(void* const* d_in, const int* in_sizes, int n_in,
                              void* d_out, int out_size, void* d_ws, size_t ws_size,
                              hipStream_t stream) {
  const float* x      = (const float*)d_in[0];
  const float* pos    = (const float*)d_in[1];
  const float* conv_w = (const float*)d_in[2];
  const float* conv_b = (const float*)d_in[3];
  const float* w1     = (const float*)d_in[4];
  const float* b1     = (const float*)d_in[5];
  const float* w2     = (const float*)d_in[6];
  const float* b2     = (const float*)d_in[7];
  float* out  = (float*)d_out;
  float* feat = (float*)d_ws;  // 4*64*128*128 floats = 16 MB scratch

  // conv5x5 + ReLU
  conv5_relu_kernel<<<16384, 256, 0, stream>>>(x, conv_w, conv_b, feat);
  // fused pos2weight GEMM (WMMA) + dynamic conv + pixel shuffle + mean add
  meta_upscale_kernel<<<4096, 128, 0, stream>>>(feat, pos, w1, b1, w2, b2, out);
}